// TVDenoiser_26972394619035
// MI455X (gfx1250) — compile-verified
//
#include <hip/hip_runtime.h>
#include <stdint.h>

// TV prox (Chambolle-Pock), 20 iterations on (4,3,512,512) f32.
// Strategy: entire state (~50MB) lives in L2 (192MB); one fused kernel per
// iteration (kernel boundary = device-wide sync for the stencil dependency);
// LDS-tiled so the primal field x is computed once per tile and reused by
// both the x2 relaxation and the dual (u) update. The x2 tile is staged
// into LDS with CDNA5 async global->LDS DMA (ASYNCcnt path).

#define TAU   0.01f
#define RHO   1.99f
#define SIGMA 12.5f          // 1/TAU/8
#define N_IT  20

#define H    512
#define W    512
#define NIMG 12              // 4*3 independent planes

#define TY   32
#define TX   64
#define LDSH (TY + 1)        // 33 (one halo row below)
#define LDSW (TX + 1)        // 65 logical cols (one halo col right)
#define LDSP 68              // physical LDS row stride in floats (272B, 16B-aligned)

__global__ __launch_bounds__(256) void tv_init_kernel(
    const float* __restrict__ y,
    float* __restrict__ x2, float* __restrict__ uh, float* __restrict__ uw, int n)
{
    int i = blockIdx.x * 256 + threadIdx.x;
    if (i < n) {
        x2[i] = y[i];
        uh[i] = 0.0f;
        uw[i] = 0.0f;
    }
}

__global__ __launch_bounds__(256) void tv_iter_kernel(
    const float* __restrict__ y,
    const float* __restrict__ x2_in,
    const float* __restrict__ uh_in,
    const float* __restrict__ uw_in,
    float* __restrict__ x2_out,
    float* __restrict__ uh_out,
    float* __restrict__ uw_out,
    const float* __restrict__ ths,
    int final_it)
{
    __shared__ __align__(16) float xs [LDSH * LDSP];  // primal x over extended tile
    __shared__ __align__(16) float x2s[LDSH * LDSP];  // old x2, DMA-staged

    const float lambd = ths[0];
    const int   i0    = blockIdx.y * TY;
    const int   j0    = blockIdx.x * TX;
    const int   base  = (int)blockIdx.z * (H * W);
    const int   tid   = threadIdx.x;

    // ---- Stage x2 extended tile into LDS via async global->LDS DMA.
    // 33 rows x (16 x b128 covering cols 0..63, + 1 x b32 for halo col 64).
    // All b128 transfers are 16B-aligned on both sides (LDSP*4 = 272 = 17*16).
    {
        const uint32_t lds0 = (uint32_t)(uintptr_t)x2s;
        for (int e = tid; e < LDSH * 17; e += 256) {
            int ii = e / 17;
            int t  = e - ii * 17;
            int gi = i0 + ii;
            if (gi < H) {
                if (t < 16) {
                    const float* gp = x2_in + (base + gi * W + j0 + 4 * t);
                    uint32_t la = lds0 + (uint32_t)(ii * LDSP + 4 * t) * 4u;
                    asm volatile("global_load_async_to_lds_b128 %0, %1, off"
                                 :: "v"(la), "v"(gp) : "memory");
                } else if (j0 + 64 < W) {
                    const float* gp = x2_in + (base + gi * W + j0 + 64);
                    uint32_t la = lds0 + (uint32_t)(ii * LDSP + 64) * 4u;
                    asm volatile("global_load_async_to_lds_b32 %0, %1, off"
                                 :: "v"(la), "v"(gp) : "memory");
                }
            }
        }
        asm volatile("s_wait_asynccnt 0x0" ::: "memory");
    }
    __syncthreads();

    // ---- Phase A: x = (x2 - TAU * div_adj(u2) + TAU * y) / (1 + TAU)
    // over the (TY+1)x(TX+1) extended tile. Boundary handling matches the
    // reference nabla_adjoint exactly:
    //   adj_h(i) = (i>0 ? uh[i-1] : 0) - (i<H-1 ? uh[i] : 0), same for w.
    for (int e = tid; e < LDSH * LDSW; e += 256) {
        int ii = e / LDSW;
        int jj = e - ii * LDSW;
        int gi = i0 + ii;
        int gj = j0 + jj;
        int le = ii * LDSP + jj;
        float xv = 0.0f;
        if (gi < H && gj < W) {
            int idx = base + gi * W + gj;
            float adj = 0.0f;
            if (gi > 0)     adj += uh_in[idx - W];
            if (gi < H - 1) adj -= uh_in[idx];
            if (gj > 0)     adj += uw_in[idx - 1];
            if (gj < W - 1) adj -= uw_in[idx];
            float x2v = x2s[le];                 // from async-staged LDS
            xv = (x2v - TAU * adj + TAU * y[idx]) / (1.0f + TAU);
        }
        xs[le] = xv;
    }
    __syncthreads();

    // ---- Phase B: interior TYxTX updates.
    for (int e = tid; e < TY * TX; e += 256) {
        int ii = e >> 6;          // / TX (TX == 64)
        int jj = e & (TX - 1);
        int gi = i0 + ii;
        int gj = j0 + jj;
        int idx = base + gi * W + gj;
        int le  = ii * LDSP + jj;

        float xc  = xs[le];
        float x2c = x2s[le];

        // x2_new = x2 + RHO * (x - x2)
        x2_out[idx] = x2c + RHO * (xc - x2c);

        if (!final_it) {
            // v = 2x - x2_old ; forward differences with zero at far edge
            float vc = 2.0f * xc - x2c;
            float gh = 0.0f, gw = 0.0f;
            if (gi < H - 1) {
                int ld = le + LDSP;
                gh = (2.0f * xs[ld] - x2s[ld]) - vc;
            }
            if (gj < W - 1) {
                int lr = le + 1;
                gw = (2.0f * xs[lr] - x2s[lr]) - vc;
            }
            float uhv = uh_in[idx];
            float uwv = uw_in[idx];
            float uth = uhv + SIGMA * gh;
            float utw = uwv + SIGMA * gw;
            float nrm = sqrtf(uth * uth + utw * utw);
            // u = ut / max(nrm/lambd, 1)  ==  ut * (lambd / max(nrm, lambd))
            float s = lambd / fmaxf(nrm, lambd);
            uh_out[idx] = uhv + RHO * (uth * s - uhv);
            uw_out[idx] = uwv + RHO * (utw * s - uwv);
        }
    }
}

extern "C" void kernel_launch(void* const* d_in, const int* in_sizes, int n_in,
                              void* d_out, int out_size, void* d_ws, size_t ws_size,
                              hipStream_t stream)
{
    (void)n_in; (void)out_size; (void)ws_size;

    const float* y   = (const float*)d_in[0];   // (4,3,512,512) f32
    const float* ths = (const float*)d_in[1];   // (1,) f32
    float*       out = (float*)d_out;

    const int N = in_sizes[0];                  // 12*512*512 = 3,145,728

    // Workspace layout: 6 ping-pong state planes (x2, uh, uw) x 2.
    float* ws  = (float*)d_ws;
    float* x2a = ws + (size_t)0 * N;
    float* x2b = ws + (size_t)1 * N;
    float* uha = ws + (size_t)2 * N;
    float* uhb = ws + (size_t)3 * N;
    float* uwa = ws + (size_t)4 * N;
    float* uwb = ws + (size_t)5 * N;

    tv_init_kernel<<<(N + 255) / 256, 256, 0, stream>>>(y, x2a, uha, uwa, N);

    dim3 grid(W / TX, H / TY, NIMG);   // (8, 16, 12)
    for (int k = 0; k < N_IT; ++k) {
        const int fin = (k == N_IT - 1);
        const float *xi, *uih, *uiw;
        float *xo, *uoh, *uow;
        if ((k & 1) == 0) { xi = x2a; uih = uha; uiw = uwa; xo = x2b; uoh = uhb; uow = uwb; }
        else              { xi = x2b; uih = uhb; uiw = uwb; xo = x2a; uoh = uha; uow = uwa; }
        if (fin) xo = out;  // last iteration writes x2 directly to output
        tv_iter_kernel<<<grid, 256, 0, stream>>>(y, xi, uih, uiw, xo, uoh, uow, ths, fin);
    }
}